// CumulativeNormalizer_77618648973444
// MI455X (gfx1250) — compile-verified
//
#include <hip/hip_runtime.h>
#include <hip/hip_bf16.h>
#include <stdint.h>

#define FRAMES   4000
#define WPB      8                      // waves per block (wave32)
#define CHUNK    256                    // frames per wave per pipeline stage
#define NCHUNK   ((FRAMES + CHUNK - 1) / CHUNK)   // 16 (15 full + 160 tail)
#define NBUF     4                      // async pipeline depth (3 chunks ahead)
#define EPSF     1e-4f

typedef __attribute__((ext_vector_type(4))) float v4f;   // native vector type
                                                         // (required by
                                                         // __builtin_nontemporal_store)

// ---------------------------------------------------------------------------
// Causal cumulative normalization, one wave32 per row.
// CDNA5 path: depth-4 GLOBAL_LOAD_ASYNC_TO_LDS_B128 pipeline + S_WAIT_ASYNCCNT,
// wave32 Hillis-Steele scan, NT stores to protect L2 from the 512 MB stream.
// ---------------------------------------------------------------------------
__global__ __launch_bounds__(WPB * 32)
void CumulativeNormalizer_77618648973444_kernel(const float* __restrict__ x,
                                                float* __restrict__ out,
                                                int rows) {
    // 8 waves * 4 buffers * 256 floats * 4 B = 32 KB LDS per block
    __shared__ __align__(16) float smem[WPB][NBUF][CHUNK];

    const int tid  = threadIdx.x;
    const int wave = tid >> 5;
    const int lane = tid & 31;
    const int row  = blockIdx.x * WPB + wave;
    if (row >= rows) return;               // no barriers used; waves independent

    const float* __restrict__ rp = x   + (size_t)row * FRAMES;
    float*       __restrict__ op = out + (size_t)row * FRAMES;

    // Low 32 bits of the flat LDS address = byte offset within the WG's LDS.
    const uint32_t ldsBase = (uint32_t)(uintptr_t)(&smem[wave][0][0]);

    // Issue one chunk: two async b128 loads, each moving 512 contiguous bytes
    // of the row into LDS in frame order. OOB lanes clamp to FRAMES-4 (their
    // contributions only affect frames past the row end -> never stored).
    auto issue = [&](int c) {
        const int base = c * CHUNK;
        int fA = base +        4 * lane; if (fA > FRAMES - 4) fA = FRAMES - 4;
        int fB = base + 128 +  4 * lane; if (fB > FRAMES - 4) fB = FRAMES - 4;
        const uint32_t bufOff = ldsBase + (uint32_t)((c & (NBUF - 1)) * (CHUNK * 4));
        asm volatile("global_load_async_to_lds_b128 %0, %1, off"
                     :: "v"(bufOff +        lane * 16u), "v"(rp + fA) : "memory");
        asm volatile("global_load_async_to_lds_b128 %0, %1, off"
                     :: "v"(bufOff + 512u + lane * 16u), "v"(rp + fB) : "memory");
    };

    float carryS = 0.0f, carryQ = 0.0f;     // running sum / sum-of-squares

    // Prime the pipeline: 3 chunks in flight before first compute.
    issue(0);
    issue(1);
    issue(2);

    for (int c = 0; c < NCHUNK; ++c) {
        // Keep 3 chunks ahead; wait until chunk c's 2 loads have landed
        // (async loads complete in order; each later chunk holds 2 counts).
        const int remaining = NCHUNK - 1 - c;
        if (remaining >= 3) {
            issue(c + 3);
            asm volatile("s_wait_asynccnt 6" ::: "memory");
        } else if (remaining == 2) {
            asm volatile("s_wait_asynccnt 4" ::: "memory");
        } else if (remaining == 1) {
            asm volatile("s_wait_asynccnt 2" ::: "memory");
        } else {
            asm volatile("s_wait_asynccnt 0" ::: "memory");
        }

        // Each lane consumes 8 *contiguous* frames from LDS -> single wave scan.
        const float* sb = &smem[wave][c & (NBUF - 1)][0];
        const v4f va = *(const v4f*)(sb + 8 * lane);
        const v4f vb = *(const v4f*)(sb + 8 * lane + 4);
        const float xs[8] = { va.x, va.y, va.z, va.w, vb.x, vb.y, vb.z, vb.w };

        // Lane-local inclusive prefix (sum and sum-of-squares).
        float ss[8], qq[8];
        ss[0] = xs[0];
        qq[0] = xs[0] * xs[0];
        #pragma unroll
        for (int j = 1; j < 8; ++j) {
            ss[j] = ss[j - 1] + xs[j];
            qq[j] = fmaf(xs[j], xs[j], qq[j - 1]);
        }

        // Wave32 Hillis-Steele inclusive scan over lane totals.
        float iS = ss[7], iQ = qq[7];
        #pragma unroll
        for (int d = 1; d < 32; d <<= 1) {
            const float tS = __shfl_up(iS, d, 32);
            const float tQ = __shfl_up(iQ, d, 32);
            if (lane >= d) { iS += tS; iQ += tQ; }
        }
        const float exS  = carryS + iS - ss[7];   // exclusive offset for this lane
        const float exQ  = carryQ + iQ - qq[7];
        const float totS = __shfl(iS, 31, 32);
        const float totQ = __shfl(iQ, 31, 32);
        carryS += totS;
        carryQ += totQ;

        // Normalize: (x - S/n)/sqrt(Q/n - S^2/n^2 + eps)
        //          = (n*x - S) * rsqrt(n*(Q + eps*n) - S^2)
        const int   f0 = c * CHUNK + 8 * lane;
        const float n0 = (float)(f0 + 1);
        float o[8];
        #pragma unroll
        for (int j = 0; j < 8; ++j) {
            const float n  = n0 + (float)j;
            const float cs = exS + ss[j];
            const float cq = exQ + qq[j];
            const float t  = fmaf(n, fmaf(EPSF, n, cq), -(cs * cs));
            o[j] = fmaf(n, xs[j], -cs) * __builtin_amdgcn_rsqf(t);
        }

        // NT stores: output is touched exactly once; don't pollute the 192 MB
        // L2 that the async read-ahead stream wants. FRAMES % 4 == 0, so
        // masking at float4 granularity is exact.
        if (f0 + 3 < FRAMES) {
            const v4f w0 = { o[0], o[1], o[2], o[3] };
            __builtin_nontemporal_store(w0, (v4f*)(op + f0));
        }
        if (f0 + 7 < FRAMES) {
            const v4f w1 = { o[4], o[5], o[6], o[7] };
            __builtin_nontemporal_store(w1, (v4f*)(op + f0 + 4));
        }
    }
}

extern "C" void kernel_launch(void* const* d_in, const int* in_sizes, int n_in,
                              void* d_out, int out_size, void* d_ws, size_t ws_size,
                              hipStream_t stream) {
    const float* x   = (const float*)d_in[0];
    float*       out = (float*)d_out;
    const int total  = in_sizes[0];            // 32*512*4000
    const int rows   = total / FRAMES;         // 16384 independent rows
    const int blocks = (rows + WPB - 1) / WPB; // 2048 blocks of 8 waves

    hipLaunchKernelGGL(CumulativeNormalizer_77618648973444_kernel,
                       dim3(blocks), dim3(WPB * 32), 0, stream,
                       x, out, rows);
}